// SOT_49469433316005
// MI455X (gfx1250) — compile-verified
//
#include <hip/hip_runtime.h>
#include <math.h>

#define B_ 8
#define N_ 2048
#define D_ 512
#define OT_REG 0.1f
#define INV_REG 10.0f
#define SINK_ITERS 10
#define THRESH 0.1f
#define DIAG_VAL 1000.0f

typedef float v2f __attribute__((ext_vector_type(2)));
typedef float v8f __attribute__((ext_vector_type(8)));

// ---------------- init: zero u, v, err, done ----------------
__global__ void k_init(float* u, float* v, float* err, int* done) {
    int t = blockIdx.x * blockDim.x + threadIdx.x;
    if (t < B_ * N_) { u[t] = 0.0f; v[t] = 0.0f; }
    if (t == 0) { *err = 0.0f; *done = 0; }
}

// ---------------- sq[b,i] = sum_d X[b,i,d]^2 (wave per row) ----------------
__global__ void k_sq(const float* __restrict__ X, float* __restrict__ sq) {
    int lane = threadIdx.x & 31;
    int row  = (blockIdx.x * blockDim.x + threadIdx.x) >> 5;   // 0..16383
    const float* r = X + (size_t)row * D_;
    float s = 0.0f;
    for (int d = lane; d < D_; d += 32) { float x = r[d]; s += x * x; }
#pragma unroll
    for (int o = 16; o > 0; o >>= 1) s += __shfl_xor(s, o, 32);
    if (lane == 0) sq[row] = s;
}

// ---- Gram via fp32 WMMA, 64x64 tile per wave (4x4 register blocking) ----
// M = sqrt(max(sq_i + sq_j - 2*X.X^T, 0))
// __launch_bounds__(256,1): allow ~170+ VGPRs so 16 v8f accumulators stay
// in registers (round 2 showed scratch spills at the default VGPR budget).
__global__ void __launch_bounds__(256, 1)
k_gram(const float* __restrict__ X, const float* __restrict__ sq,
       float* __restrict__ M) {
    int lane = threadIdx.x & 31;
    int wave = (blockIdx.x * blockDim.x + threadIdx.x) >> 5;   // 0..8191
    const int TPB = (N_ / 64) * (N_ / 64);                     // 1024 tiles/batch
    int b  = wave / TPB;
    int t  = wave % TPB;
    int ib = (t / (N_ / 64)) * 64;
    int jb = (t % (N_ / 64)) * 64;
    int m  = lane & 15;                              // row/col within 16-tile
    int kg = lane >> 4;                              // K-pair group (0/1)

    // A-fragment layout (16x4 fp32): lane m holds M=m; VGPR v holds K=2*kg+v
    const float* Abase = X + ((size_t)b * N_ + ib + m) * D_ + 2 * kg;
    const float* Bbase = X + ((size_t)b * N_ + jb + m) * D_ + 2 * kg;

    v8f acc[4][4];
#pragma unroll
    for (int ti = 0; ti < 4; ++ti)
#pragma unroll
        for (int tj = 0; tj < 4; ++tj)
            acc[ti][tj] = (v8f){};

#pragma unroll 1
    for (int k = 0; k < D_; k += 4) {
        v2f a[4], bb[4];
#pragma unroll
        for (int ti = 0; ti < 4; ++ti)
            a[ti] = *(const v2f*)(Abase + (size_t)ti * 16 * D_ + k);
#pragma unroll
        for (int tj = 0; tj < 4; ++tj)
            bb[tj] = *(const v2f*)(Bbase + (size_t)tj * 16 * D_ + k);
#pragma unroll
        for (int ti = 0; ti < 4; ++ti)
#pragma unroll
            for (int tj = 0; tj < 4; ++tj)
                acc[ti][tj] = __builtin_amdgcn_wmma_f32_16x16x4_f32(
                    false, a[ti], false, bb[tj], (short)0, acc[ti][tj],
                    false, false);
    }

    // C/D layout: lane -> N = lane&15; VGPR r -> M = r + 8*(lane>>4)
    int rbase = kg * 8;
#pragma unroll
    for (int ti = 0; ti < 4; ++ti) {
        float sqi[8];
#pragma unroll
        for (int r = 0; r < 8; ++r)
            sqi[r] = sq[(size_t)b * N_ + ib + ti * 16 + rbase + r];
#pragma unroll
        for (int tj = 0; tj < 4; ++tj) {
            float sqj = sq[(size_t)b * N_ + jb + tj * 16 + m];
#pragma unroll
            for (int r = 0; r < 8; ++r) {
                int i = ib + ti * 16 + rbase + r;
                int j = jb + tj * 16 + m;
                float d2 = sqi[r] + sqj - 2.0f * acc[ti][tj][r];
                M[((size_t)b * N_ + i) * N_ + j] = sqrtf(fmaxf(d2, 0.0f));
            }
        }
    }
}

// ---------------- row max (wave per row) ----------------
__global__ void k_rowmax(const float* __restrict__ M, float* __restrict__ rowmax) {
    int lane = threadIdx.x & 31;
    int row  = (blockIdx.x * blockDim.x + threadIdx.x) >> 5;
    const float* Mr = M + (size_t)row * N_;
    float mx = 0.0f;
    for (int j = lane; j < N_; j += 32) mx = fmaxf(mx, Mr[j]);
#pragma unroll
    for (int o = 16; o > 0; o >>= 1) mx = fmaxf(mx, __shfl_xor(mx, o, 32));
    if (lane == 0) rowmax[row] = mx;
}

// ---------------- normalize rows, set diag = DIAG_VAL ----------------
__global__ void k_norm(float* __restrict__ M, const float* __restrict__ rowmax) {
    size_t tot    = (size_t)B_ * N_ * N_;
    size_t stride = (size_t)gridDim.x * blockDim.x;
    for (size_t idx = (size_t)blockIdx.x * blockDim.x + threadIdx.x; idx < tot; idx += stride) {
        size_t row = idx / N_;
        int j = (int)(idx % N_);
        int i = (int)(row % N_);
        M[idx] = (i == j) ? DIAG_VAL : M[idx] / rowmax[row];
    }
}

// ---------------- Sinkhorn u update (wave per row, online LSE) --------------
__global__ void k_u(const float* __restrict__ M, float* __restrict__ u,
                    const float* __restrict__ v, float* err, const int* done,
                    float log_mu) {
    if (*done) return;
    int lane = threadIdx.x & 31;
    int row  = (blockIdx.x * blockDim.x + threadIdx.x) >> 5;   // 0..16383
    int b    = row / N_;
    const float* Mr = M + (size_t)row * N_;
    const float* vb = v + (size_t)b * N_;
    float ui = u[row];
    float mx = -INFINITY, s = 0.0f;
    for (int j = lane; j < N_; j += 32) {
        float t  = (ui + vb[j] - Mr[j]) * INV_REG;
        float nm = fmaxf(mx, t);
        s  = s * __expf(mx - nm) + __expf(t - nm);
        mx = nm;
    }
#pragma unroll
    for (int o = 16; o > 0; o >>= 1) {
        float om = __shfl_xor(mx, o, 32);
        float os = __shfl_xor(s,  o, 32);
        float nm = fmaxf(mx, om);
        s  = s * __expf(mx - nm) + os * __expf(om - nm);
        mx = nm;
    }
    float lse = mx + __logf(s);
    float un  = OT_REG * (log_mu - lse) + ui;
    if (lane == 0) {
        atomicAdd(err, fabsf(un - ui));
        u[row] = un;
    }
}

// ---------------- Sinkhorn v update (thread per column, coalesced) ----------
__global__ void k_v(const float* __restrict__ M, const float* __restrict__ u,
                    float* __restrict__ v, const int* done, float log_nu) {
    if (*done) return;
    int gt = blockIdx.x * blockDim.x + threadIdx.x;            // 0..16383
    int b  = gt / N_;
    int j  = gt % N_;
    const float* Mc = M + (size_t)b * N_ * N_ + j;
    const float* ub = u + (size_t)b * N_;
    float vj = v[gt];
    float mx = -INFINITY, s = 0.0f;
    for (int i = 0; i < N_; ++i) {
        float t  = (ub[i] + vj - Mc[(size_t)i * N_]) * INV_REG;
        float nm = fmaxf(mx, t);
        s  = s * __expf(mx - nm) + __expf(t - nm);
        mx = nm;
    }
    float lse = mx + __logf(s);
    v[gt] = OT_REG * (log_nu - lse) + vj;
}

// ---------------- convergence flag update ----------------
__global__ void k_done(float* err, int* done) {
    if (!*done && (*err / (float)B_) < THRESH) *done = 1;
    *err = 0.0f;
}

// ---------------- per-batch max of log_p, stage 1 ----------------
__global__ void k_bmax(const float* __restrict__ M, const float* __restrict__ u,
                       const float* __restrict__ v, float* __restrict__ partial) {
    __shared__ float sm[256];
    int b   = blockIdx.x >> 6;                  // 64 blocks per batch
    int sub = blockIdx.x & 63;
    int r0  = sub * (N_ / 64);                  // 32 rows each
    const int elems = (N_ / 64) * N_;
    float mx = -INFINITY;
    for (int idx = threadIdx.x; idx < elems; idx += blockDim.x) {
        int i = r0 + idx / N_;
        int j = idx % N_;
        float lp = (u[b * N_ + i] + v[b * N_ + j] -
                    M[((size_t)b * N_ + i) * N_ + j]) * INV_REG;
        mx = fmaxf(mx, lp);
    }
    sm[threadIdx.x] = mx;
    __syncthreads();
    for (int o = 128; o > 0; o >>= 1) {
        if (threadIdx.x < o) sm[threadIdx.x] = fmaxf(sm[threadIdx.x], sm[threadIdx.x + o]);
        __syncthreads();
    }
    if (threadIdx.x == 0) partial[blockIdx.x] = sm[0];
}

// ---------------- per-batch max, stage 2 ----------------
__global__ void k_bmax2(const float* __restrict__ partial, float* __restrict__ maxlp) {
    int b    = blockIdx.x;
    int lane = threadIdx.x;                     // 32 threads
    float mx = fmaxf(partial[b * 64 + lane], partial[b * 64 + 32 + lane]);
#pragma unroll
    for (int o = 16; o > 0; o >>= 1) mx = fmaxf(mx, __shfl_xor(mx, o, 32));
    if (lane == 0) maxlp[b] = mx;
}

// ---------------- final: features = sigmoid(log_p)/sigmoid(max), diag=1 -----
__global__ void k_final(float* __restrict__ M, const float* __restrict__ u,
                        const float* __restrict__ v, const float* __restrict__ maxlp) {
    size_t tot    = (size_t)B_ * N_ * N_;
    size_t stride = (size_t)gridDim.x * blockDim.x;
    for (size_t idx = (size_t)blockIdx.x * blockDim.x + threadIdx.x; idx < tot; idx += stride) {
        size_t row = idx / N_;
        int j = (int)(idx % N_);
        int b = (int)(row / N_);
        int i = (int)(row % N_);
        float out;
        if (i == j) {
            out = 1.0f;
        } else {
            float lp = (u[row] + v[(size_t)b * N_ + j] - M[idx]) * INV_REG;
            float f  = 1.0f / (1.0f + __expf(-lp));
            float fm = 1.0f / (1.0f + __expf(-maxlp[b]));
            out = f / fm;
        }
        M[idx] = out;
    }
}

extern "C" void kernel_launch(void* const* d_in, const int* in_sizes, int n_in,
                              void* d_out, int out_size, void* d_ws, size_t ws_size,
                              hipStream_t stream) {
    (void)in_sizes; (void)n_in; (void)out_size; (void)ws_size;
    const float* X = (const float*)d_in[0];
    float* M = (float*)d_out;                   // M lives in d_out (128 MiB, L2-resident)

    // small workspace layout
    float* sq      = (float*)d_ws;              // 16384
    float* rowmax  = sq + B_ * N_;              // 16384
    float* u       = rowmax + B_ * N_;          // 16384
    float* v       = u + B_ * N_;               // 16384
    float* partial = v + B_ * N_;               // 512
    float* maxlp   = partial + B_ * 64;         // 8
    float* err     = maxlp + B_;                // 1
    int*   done    = (int*)(err + 1);           // 1

    const float log_mu = logf(1.0f / (float)N_ + 1e-8f);

    k_init  <<<64,    256, 0, stream>>>(u, v, err, done);
    k_sq    <<<2048,  256, 0, stream>>>(X, sq);
    k_gram  <<<1024,  256, 0, stream>>>(X, sq, M);      // 8192 waves, 64x64 tile each
    k_rowmax<<<2048,  256, 0, stream>>>(M, rowmax);
    k_norm  <<<16384, 256, 0, stream>>>(M, rowmax);

    for (int it = 0; it < SINK_ITERS; ++it) {
        k_u   <<<2048, 256, 0, stream>>>(M, u, v, err, done, log_mu);
        k_v   <<<64,   256, 0, stream>>>(M, u, v, done, log_mu);
        k_done<<<1,    1,   0, stream>>>(err, done);
    }

    k_bmax  <<<B_ * 64, 256, 0, stream>>>(M, u, v, partial);
    k_bmax2 <<<B_,      32,  0, stream>>>(partial, maxlp);
    k_final <<<16384,   256, 0, stream>>>(M, u, v, maxlp);
}